// Proposal_layer_32143535243431
// MI455X (gfx1250) — compile-verified
//
#include <hip/hip_runtime.h>
#include <hip/hip_bf16.h>
#include <math.h>

// ---------------------------------------------------------------------------
// RPN proposal layer for MI455X (gfx1250).
// - Bandwidth-bound decode: no matmul in this op -> WMMA not applicable.
//   gfx1250-specific path used instead: async global->LDS DMA (ASYNCcnt),
//   double-buffered per wave (wait 0x2 -> process tile0 while tile1 streams).
// - top-6000 via histogram threshold + 8192-wide LDS bitonic sort (64 KB of
//   the WGP's 320 KB LDS in a single workgroup).
// - greedy single-WG NMS == reference's argmax-scan semantics.
// - If ws_size allows, pass 1 caches masked scores (9.4 MB) so pass 2 reads
//   9.4 MB instead of re-streaming the 56.6 MB inputs (~35% less traffic).
// ---------------------------------------------------------------------------

#define MAP_W   512
#define MAP_H   512
#define NA      9
#define NTOT    (MAP_W * MAP_H * NA)   // 2,359,296
#define NBINS   4096
#define CAP     8192
#define PRE     6000
#define POST    300
#define IMG_MAX 8191.0f
#define MIN_SZ  16.0f
#define NEG_INF (-__builtin_inff())

// Base anchor widths/heights from _generate_anchors(16,(.5,1,2),(8,16,32)).
__device__ __constant__ float c_bw[NA] = {184.f, 368.f, 736.f, 128.f, 256.f, 512.f,  88.f, 176.f, 352.f};
__device__ __constant__ float c_bh[NA] = { 96.f, 192.f, 384.f, 128.f, 256.f, 512.f, 176.f, 352.f, 704.f};

struct BoxV { float x1, y1, x2, y2; int valid; };

// anchor center is exactly (16x+8, 16y+8); width/height = base w/h.
__device__ __forceinline__ BoxV decode_box(int cell, int a,
                                           float d0, float d1, float d2, float d3) {
    int x = cell & (MAP_W - 1);
    int y = cell >> 9;
    float aw  = c_bw[a], ah = c_bh[a];
    float acx = (float)x * 16.0f + 8.0f;
    float acy = (float)y * 16.0f + 8.0f;
    float pcx = d0 * aw + acx;
    float pcy = d1 * ah + acy;
    float pw  = expf(d2) * aw;
    float ph  = expf(d3) * ah;
    float x1 = pcx - 0.5f * pw, x2 = pcx + 0.5f * pw;
    float y1 = pcy - 0.5f * ph, y2 = pcy + 0.5f * ph;
    x1 = fminf(fmaxf(x1, 0.0f), IMG_MAX);
    x2 = fminf(fmaxf(x2, 0.0f), IMG_MAX);
    y1 = fminf(fmaxf(y1, 0.0f), IMG_MAX);
    y2 = fminf(fmaxf(y2, 0.0f), IMG_MAX);
    BoxV b;
    b.x1 = x1; b.y1 = y1; b.x2 = x2; b.y2 = y2;
    b.valid = (x2 - x1 + 1.0f >= MIN_SZ) && (y2 - y1 + 1.0f >= MIN_SZ);
    return b;
}

// wave-private async stage of one anchor's 4 deltas + 1 score into LDS.
__device__ __forceinline__ void stage_async(const float* gd, const float* gs,
                                            float* sd_slot, float* ss_slot) {
    unsigned lds_d = (unsigned)(unsigned long long)sd_slot;
    unsigned lds_s = (unsigned)(unsigned long long)ss_slot;
    asm volatile("global_load_async_to_lds_b128 %0, %1, off"
                 :: "v"(lds_d), "v"((unsigned long long)gd) : "memory");
    asm volatile("global_load_async_to_lds_b32 %0, %1, off"
                 :: "v"(lds_s), "v"((unsigned long long)gs) : "memory");
}

// -------------------------------------------------------------------- init --
__global__ __launch_bounds__(1024) void k_init(int* __restrict__ hist,
                                               int* __restrict__ cnt,
                                               float* __restrict__ out) {
    int t = threadIdx.x;
    for (int b = t; b < NBINS; b += 1024) hist[b] = 0;
    if (t == 0) cnt[0] = 0;
    for (int i = t; i < POST * 5; i += 1024) out[i] = 0.0f;  // 1200 boxes + 300 scores
}

// ----------------- pass 1: decode + histogram (double-buffered async DMA) ---
__global__ __launch_bounds__(256) void k_decode_hist(const float* __restrict__ delta,
                                                     const float* __restrict__ score,
                                                     int* __restrict__ hist,
                                                     float* __restrict__ sc,
                                                     int use_cache) {
    __shared__ float sd[2][256 * 4];
    __shared__ float ss[2][256];
    __shared__ int   lh[NBINS];
    int tid = threadIdx.x;
    for (int b = tid; b < NBINS; b += 256) lh[b] = 0;

    int i0 = blockIdx.x * 512 + tid;   // grid sized exactly NTOT/512 blocks
    int i1 = i0 + 256;

    // issue both tiles' DMA up front (4 async ops outstanding per lane/wave)
    {
        int c0 = i0 / NA, a0 = i0 - c0 * NA;
        int c1 = i1 / NA, a1 = i1 - c1 * NA;
        stage_async(delta + (size_t)c0 * 36 + a0 * 4,
                    score + (size_t)c0 * 18 + NA + a0,
                    &sd[0][tid * 4], &ss[0][tid]);
        stage_async(delta + (size_t)c1 * 36 + a1 * 4,
                    score + (size_t)c1 * 18 + NA + a1,
                    &sd[1][tid * 4], &ss[1][tid]);
    }
    __syncthreads();   // lh initialized before any atomics below

#pragma unroll
    for (int t = 0; t < 2; ++t) {
        // async loads complete in order: <=2 outstanding => tile0's pair done
        if (t == 0) asm volatile("s_wait_asynccnt 0x2" ::: "memory");
        else        asm volatile("s_wait_asynccnt 0x0" ::: "memory");
        int i    = (t == 0) ? i0 : i1;
        int cell = i / NA;
        int a    = i - cell * NA;
        float d0 = sd[t][tid * 4 + 0], d1 = sd[t][tid * 4 + 1];
        float d2 = sd[t][tid * 4 + 2], d3 = sd[t][tid * 4 + 3];
        float s  = ss[t][tid];
        BoxV bx = decode_box(cell, a, d0, d1, d2, d3);
        float masked = bx.valid ? s : NEG_INF;
        if (use_cache) sc[i] = masked;
        if (bx.valid) {
            int b = (int)(s * (float)NBINS);
            b = max(0, min(NBINS - 1, b));
            atomicAdd(&lh[b], 1);
        }
    }
    __syncthreads();
    for (int b = tid; b < NBINS; b += 256) {
        int v = lh[b];
        if (v) atomicAdd(&hist[b], v);
    }
}

// ----------------------------------------------------- threshold selection --
__global__ void k_thresh(const int* __restrict__ hist, float* __restrict__ thr) {
    if (threadIdx.x == 0 && blockIdx.x == 0) {
        int cum = 0;
        float t = 0.0f;
        for (int b = NBINS - 1; b >= 0; --b) {
            cum += hist[b];
            if (cum >= PRE) { t = (float)b * (1.0f / (float)NBINS); break; }
        }
        thr[0] = t;
    }
}

// ------------- pass 2a (fallback): re-decode + compact, async pipelined -----
__global__ __launch_bounds__(256) void k_decode_compact(const float* __restrict__ delta,
                                                        const float* __restrict__ score,
                                                        const float* __restrict__ thr,
                                                        int* __restrict__ cnt,
                                                        float* __restrict__ cs,
                                                        int* __restrict__ ci) {
    __shared__ float sd[2][256 * 4];
    __shared__ float ss[2][256];
    int tid = threadIdx.x;
    int i0  = blockIdx.x * 512 + tid;
    int i1  = i0 + 256;
    {
        int c0 = i0 / NA, a0 = i0 - c0 * NA;
        int c1 = i1 / NA, a1 = i1 - c1 * NA;
        stage_async(delta + (size_t)c0 * 36 + a0 * 4,
                    score + (size_t)c0 * 18 + NA + a0,
                    &sd[0][tid * 4], &ss[0][tid]);
        stage_async(delta + (size_t)c1 * 36 + a1 * 4,
                    score + (size_t)c1 * 18 + NA + a1,
                    &sd[1][tid * 4], &ss[1][tid]);
    }
    float t0 = thr[0];
#pragma unroll
    for (int t = 0; t < 2; ++t) {
        if (t == 0) asm volatile("s_wait_asynccnt 0x2" ::: "memory");
        else        asm volatile("s_wait_asynccnt 0x0" ::: "memory");
        int i    = (t == 0) ? i0 : i1;
        int cell = i / NA;
        int a    = i - cell * NA;
        float d0 = sd[t][tid * 4 + 0], d1 = sd[t][tid * 4 + 1];
        float d2 = sd[t][tid * 4 + 2], d3 = sd[t][tid * 4 + 3];
        float s  = ss[t][tid];
        BoxV bx = decode_box(cell, a, d0, d1, d2, d3);
        if (bx.valid && s >= t0) {
            int pos = atomicAdd(cnt, 1);
            if (pos < CAP) { cs[pos] = s; ci[pos] = i; }
        }
    }
}

// ------------- pass 2b (cached): compact from 9.4 MB score array ------------
__global__ __launch_bounds__(256) void k_compact_cached(const float4* __restrict__ sc4,
                                                        const float* __restrict__ thr,
                                                        int* __restrict__ cnt,
                                                        float* __restrict__ cs,
                                                        int* __restrict__ ci) {
    int q = blockIdx.x * 256 + threadIdx.x;   // grid covers NTOT/4 exactly
    float t = thr[0];
    float4 v = sc4[q];
    int base = q * 4;
    if (v.x >= t) { int p = atomicAdd(cnt, 1); if (p < CAP) { cs[p] = v.x; ci[p] = base + 0; } }
    if (v.y >= t) { int p = atomicAdd(cnt, 1); if (p < CAP) { cs[p] = v.y; ci[p] = base + 1; } }
    if (v.z >= t) { int p = atomicAdd(cnt, 1); if (p < CAP) { cs[p] = v.z; ci[p] = base + 2; } }
    if (v.w >= t) { int p = atomicAdd(cnt, 1); if (p < CAP) { cs[p] = v.w; ci[p] = base + 3; } }
}

// -------------------------------- single-WG bitonic sort of <=8192 in LDS ---
__device__ __forceinline__ bool before_kv(float sa, int ia, float sb, int ib) {
    // desired order: score descending, index ascending on ties (matches top_k)
    return (sa > sb) || (sa == sb && ia < ib);
}

__global__ __launch_bounds__(1024) void k_sort(const int* __restrict__ cnt,
                                               const float* __restrict__ cs,
                                               const int* __restrict__ ci,
                                               float* __restrict__ ts,
                                               int* __restrict__ ti) {
    __shared__ float lsc[CAP];   // 32 KB
    __shared__ int   lid[CAP];   // 32 KB  (WGP has 320 KB LDS)
    int tid = threadIdx.x;
    int n   = min(cnt[0], CAP);
    for (int i = tid; i < CAP; i += 1024) {
        if (i < n) { lsc[i] = cs[i]; lid[i] = ci[i]; }
        else       { lsc[i] = NEG_INF; lid[i] = 0x7fffffff; }
    }
    __syncthreads();
    for (int k = 2; k <= CAP; k <<= 1) {
        for (int j = k >> 1; j > 0; j >>= 1) {
            for (int i = tid; i < CAP; i += 1024) {
                int p = i ^ j;
                if (p > i) {
                    bool up = ((i & k) == 0);
                    float sa = lsc[i]; int ia = lid[i];
                    float sb = lsc[p]; int ib = lid[p];
                    if (before_kv(sb, ib, sa, ia) == up) {
                        lsc[i] = sb; lid[i] = ib;
                        lsc[p] = sa; lid[p] = ia;
                    }
                }
            }
            __syncthreads();
        }
    }
    for (int i = tid; i < PRE; i += 1024) { ts[i] = lsc[i]; ti[i] = lid[i]; }
}

// ----------------------------------- re-decode boxes for top-PRE survivors --
__global__ __launch_bounds__(256) void k_gather(const float* __restrict__ delta,
                                                const float* __restrict__ ts,
                                                const int* __restrict__ ti,
                                                float4* __restrict__ tb) {
    int t = blockIdx.x * 256 + threadIdx.x;
    if (t >= PRE) return;
    float s  = ts[t];
    int  idx = ti[t];
    float4 b = make_float4(0.f, 0.f, 0.f, 0.f);
    if (s > NEG_INF && idx >= 0 && idx < NTOT) {
        int cell = idx / NA;
        int a    = idx - cell * NA;
        const float* gd = delta + (size_t)cell * 36 + a * 4;
        BoxV bx = decode_box(cell, a, gd[0], gd[1], gd[2], gd[3]);
        b = make_float4(bx.x1, bx.y1, bx.x2, bx.y2);
    }
    tb[t] = b;
}

// ---------------------------------------------- single-WG greedy NMS (300) --
__global__ __launch_bounds__(1024) void k_nms(const float4* __restrict__ tb,
                                              const float* __restrict__ ts,
                                              float* __restrict__ out) {
    __shared__ unsigned char sup[PRE];
    __shared__ int s_pick;
    __shared__ int s_ptr;
    int tid = threadIdx.x;
    for (int i = tid; i < PRE; i += 1024) sup[i] = 0;
    if (tid == 0) s_ptr = 0;
    __syncthreads();

    for (int slot = 0; slot < POST; ++slot) {
        if (tid == 0) {
            int p = s_ptr;
            while (p < PRE && sup[p]) ++p;
            if (p < PRE && ts[p] > NEG_INF) {
                s_pick = p;
                sup[p] = 1;
                s_ptr  = p + 1;
                float4 b = tb[p];
                out[slot * 4 + 0] = b.x;
                out[slot * 4 + 1] = b.y;
                out[slot * 4 + 2] = b.z;
                out[slot * 4 + 3] = b.w;
                out[POST * 4 + slot] = ts[p];
            } else {
                s_pick = -1;
            }
        }
        __syncthreads();
        int p = s_pick;
        if (p < 0) break;  // uniform across the block
        float4 b = tb[p];
        float barea = (b.z - b.x) * (b.w - b.y);
        for (int j = p + 1 + tid; j < PRE; j += 1024) {
            if (!sup[j]) {
                float4 c = tb[j];
                float xx1 = fmaxf(b.x, c.x), yy1 = fmaxf(b.y, c.y);
                float xx2 = fminf(b.z, c.z), yy2 = fminf(b.w, c.w);
                float inter = fmaxf(xx2 - xx1, 0.0f) * fmaxf(yy2 - yy1, 0.0f);
                float area  = (c.z - c.x) * (c.w - c.y);
                float iou   = inter / (area + barea - inter);
                if (iou > 0.7f) sup[j] = 1;
            }
        }
        __syncthreads();
    }
}

// ---------------------------------------------------------------------------
extern "C" void kernel_launch(void* const* d_in, const int* in_sizes, int n_in,
                              void* d_out, int out_size, void* d_ws, size_t ws_size,
                              hipStream_t stream) {
    const float* delta = (const float*)d_in[0];  // (1,512,512,36)
    const float* score = (const float*)d_in[1];  // (1,512,512,18)
    float* out = (float*)d_out;                  // 1200 box floats + 300 scores

    // scratch layout (all 16B aligned)
    char* w0 = (char*)d_ws;
    char* w  = w0;
    int*    hist = (int*)w;    w += NBINS * sizeof(int);
    int*    cnt  = (int*)w;    w += 16;
    float*  thr  = (float*)w;  w += 16;
    float*  cs   = (float*)w;  w += CAP * sizeof(float);
    int*    ci   = (int*)w;    w += CAP * sizeof(int);
    float*  ts   = (float*)w;  w += PRE * sizeof(float);   // 24000 B (16B mult)
    int*    ti   = (int*)w;    w += PRE * sizeof(int);     // 24000 B
    float4* tb   = (float4*)w; w += PRE * sizeof(float4);
    float*  sc   = (float*)w;  // optional NTOT-float masked-score cache
    size_t need_cache = (size_t)(w - w0) + (size_t)NTOT * sizeof(float);
    int use_cache = (ws_size >= need_cache) ? 1 : 0;

    k_init<<<1, 1024, 0, stream>>>(hist, cnt, out);
    k_decode_hist<<<NTOT / 512, 256, 0, stream>>>(delta, score, hist, sc, use_cache);
    k_thresh<<<1, 64, 0, stream>>>(hist, thr);
    if (use_cache) {
        k_compact_cached<<<NTOT / 4 / 256, 256, 0, stream>>>((const float4*)sc, thr, cnt, cs, ci);
    } else {
        k_decode_compact<<<NTOT / 512, 256, 0, stream>>>(delta, score, thr, cnt, cs, ci);
    }
    k_sort<<<1, 1024, 0, stream>>>(cnt, cs, ci, ts, ti);
    k_gather<<<(PRE + 255) / 256, 256, 0, stream>>>(delta, ts, ti, tb);
    k_nms<<<1, 1024, 0, stream>>>(tb, ts, out);
}